// CacheOnlyAttentionLayer_78924319031347
// MI455X (gfx1250) — compile-verified
//
#include <hip/hip_runtime.h>

// Problem constants (from the reference)
#define NUM_KV_HEADS 8
#define HEAD_SIZE    128
#define KV_BLOCK     16          // BLOCK_SIZE in reference (tokens per cache block)
#define NUM_BLOCKS   4096
#define TOTAL_TOKENS 32768
#define HIDDEN       (2 * NUM_KV_HEADS * HEAD_SIZE)          // 2048 floats per token
#define SLAB         (NUM_KV_HEADS * HEAD_SIZE)              // 1024 floats per (plane,slot)
#define PLANE_ELEMS  ((size_t)NUM_BLOCKS * KV_BLOCK * SLAB)  // 67,108,864 floats per plane

typedef float v4f __attribute__((ext_vector_type(4)));

// ---------------------------------------------------------------------------
// Kernel 1: stream the old cache into d_out.
// 512 MB read + 512 MB write, never re-read on device -> non-temporal b128
// loads/stores (stream is 2.7x the 192 MB L2; RT would just thrash it).
// Each thread moves 4 float4 (64 B), issued as 4 independent NT loads before
// the 4 NT stores: 2 KB outstanding per wave -> ~16 MB device-wide MLP,
// comfortably covering HBM latency at 23.3 TB/s. Accesses stride blockDim
// so every load/store instruction stays a fully coalesced 512 B/wave request.
// ---------------------------------------------------------------------------
#define COPY_TPB   256
#define COPY_ITEMS 4    // float4 per thread

__global__ __launch_bounds__(COPY_TPB) void cache_stream_copy(
    const v4f* __restrict__ src, v4f* __restrict__ dst, long long n4)
{
    const size_t base = (size_t)blockIdx.x * (COPY_TPB * COPY_ITEMS) + threadIdx.x;

    if ((long long)(base + 3 * COPY_TPB) < n4) {
        // Fast path: all 4 in range (always taken for the 2^25-float4 cache).
        v4f a = __builtin_nontemporal_load(&src[base + 0 * COPY_TPB]);
        v4f b = __builtin_nontemporal_load(&src[base + 1 * COPY_TPB]);
        v4f c = __builtin_nontemporal_load(&src[base + 2 * COPY_TPB]);
        v4f d = __builtin_nontemporal_load(&src[base + 3 * COPY_TPB]);
        __builtin_nontemporal_store(a, &dst[base + 0 * COPY_TPB]);
        __builtin_nontemporal_store(b, &dst[base + 1 * COPY_TPB]);
        __builtin_nontemporal_store(c, &dst[base + 2 * COPY_TPB]);
        __builtin_nontemporal_store(d, &dst[base + 3 * COPY_TPB]);
    } else {
        #pragma unroll
        for (int k = 0; k < COPY_ITEMS; ++k) {
            size_t i = base + (size_t)k * COPY_TPB;
            if ((long long)i < n4) {
                v4f v = __builtin_nontemporal_load(&src[i]);
                __builtin_nontemporal_store(v, &dst[i]);
            }
        }
    }
}

// ---------------------------------------------------------------------------
// Kernel 2: scatter the last token of each request into its cache slot.
// One block per request. Each thread moves one float4 of K and one of V:
//   key = hidden[last, 0:1024]    -> out[slot*1024]          (plane 0)
//   val = hidden[last, 1024:2048] -> out[PLANE + slot*1024]  (plane 1)
// slot*1024 == ((blk*16 + off) * 8 * 128) matches kv_cache[plane, blk, off].
// ---------------------------------------------------------------------------
__global__ __launch_bounds__(256) void kv_scatter_last(
    const float* __restrict__ hidden,
    const int*   __restrict__ query_start_loc,
    const int*   __restrict__ slot_mapping,
    const int*   __restrict__ num_reqs_ptr,
    float*       __restrict__ out)
{
    const int r = blockIdx.x;
    if (r >= *num_reqs_ptr) return;

    int last = query_start_loc[r + 1] - 1;           // uniform -> scalar loads
    last = last < 0 ? 0 : last;
    last = last > (TOTAL_TOKENS - 1) ? (TOTAL_TOKENS - 1) : last;
    const int slot = slot_mapping[last];

    const int t = threadIdx.x;                       // 0..255
    const v4f* __restrict__ srow =
        (const v4f*)(hidden + (size_t)last * HIDDEN);        // 512 float4 per row
    v4f* __restrict__ dkey = (v4f*)(out + (size_t)slot * SLAB);
    v4f* __restrict__ dval = (v4f*)(out + PLANE_ELEMS + (size_t)slot * SLAB);

    // 256 lanes x 16 B = 4 KB = one 1024-float slab per half.
    v4f k = srow[t];
    v4f v = srow[t + 256];
    __builtin_nontemporal_store(k, &dkey[t]);
    __builtin_nontemporal_store(v, &dval[t]);
}

// ---------------------------------------------------------------------------
// Launch: copy first, then scatter (same stream -> ordered).
// ---------------------------------------------------------------------------
extern "C" void kernel_launch(void* const* d_in, const int* in_sizes, int n_in,
                              void* d_out, int out_size, void* d_ws, size_t ws_size,
                              hipStream_t stream)
{
    const float* hidden      = (const float*)d_in[0];
    const float* kv_cache_in = (const float*)d_in[1];
    const int*   qsl         = (const int*)  d_in[2];
    const int*   slot_map    = (const int*)  d_in[3];
    const int*   num_reqs    = (const int*)  d_in[4];
    float*       out         = (float*)d_out;

    // Bulk copy: out_size = 134,217,728 floats -> 33,554,432 float4,
    // 1024 float4 per block -> 32,768 blocks (8 wave32 each).
    const long long n4 = (long long)out_size / 4;
    const long long per_block = COPY_TPB * COPY_ITEMS;
    const unsigned  copy_blocks = (unsigned)((n4 + per_block - 1) / per_block);
    cache_stream_copy<<<copy_blocks, COPY_TPB, 0, stream>>>(
        (const v4f*)kv_cache_in, (v4f*)out, n4);

    // Scatter: grid from host-known query_start_loc length (num_reqs + 1),
    // guarded on device against the actual *num_reqs value.
    const int nreq_grid = in_sizes[2] - 1;
    kv_scatter_last<<<nreq_grid, 256, 0, stream>>>(
        hidden, qsl, slot_map, num_reqs, out);
}